// NeuralVoxelField_74809740362345
// MI455X (gfx1250) — compile-verified
//
#include <hip/hip_runtime.h>

// Trilinear interpolation of a 128^3 x 16 f32 grid at N random points.
// Memory-bound gather kernel: the 128 MB grid is fully L2-resident on MI455X
// (192 MB L2), so we keep grid loads at default (regular-temporal) policy and
// stream points/output with non-temporal hints to avoid evicting it.
//
// Layout: 16 lanes (= half of a wave32) handle one point, one channel per
// lane. Each of the 8 corner gathers is a coalesced 64B request per point.

#define GD 128   // grid dim (D == H == W)
#define GC 16    // channels
static constexpr float INV_VOXEL = 20.0f;  // 1 / 0.05

__global__ __launch_bounds__(256) void trilerp16_kernel(
    const float* __restrict__ pts,   // (N, 3) f32
    const float* __restrict__ grid,  // (128,128,128,16) f32
    float* __restrict__ out,         // (N, 16) f32
    int npts)
{
    __shared__ float sc[16 * 3];

    const int tid  = threadIdx.x;
    const int base = blockIdx.x * 16;          // first point of this block

    // Cooperative, non-temporal load of 16 points' coordinates (48 floats).
    if (tid < 48) {
        const int idx = base * 3 + tid;
        sc[tid] = (idx < npts * 3) ? __builtin_nontemporal_load(&pts[idx]) : 0.0f;
    }
    // Speculative prefetch of points a few blocks ahead (global_prefetch_b8;
    // translation failures are silently dropped, so no OOB guard needed).
    if (tid == 0) {
        __builtin_prefetch(&pts[(size_t)(base + 16 * 8) * 3], 0, 0);
    }
    __syncthreads();

    const int lp = tid >> 4;     // local point 0..15
    const int c  = tid & 15;     // channel 0..15
    const int p  = base + lp;
    if (p >= npts) return;

    // Grid-space coordinates (broadcast read within half-wave: conflict-free).
    const float x = sc[lp * 3 + 0] * INV_VOXEL;
    const float y = sc[lp * 3 + 1] * INV_VOXEL;
    const float z = sc[lp * 3 + 2] * INV_VOXEL;

    int x0 = (int)floorf(x); x0 = min(max(x0, 0), GD - 1);
    int y0 = (int)floorf(y); y0 = min(max(y0, 0), GD - 1);
    int z0 = (int)floorf(z); z0 = min(max(z0, 0), GD - 1);
    const int x1 = min(x0 + 1, GD - 1);
    const int y1 = min(y0 + 1, GD - 1);
    const int z1 = min(z0 + 1, GD - 1);

    // Fractions relative to the *clipped* floor (matches reference).
    const float xd = x - (float)x0;
    const float yd = y - (float)y0;
    const float zd = z - (float)z0;

    // Element offsets: ((x*128 + y)*128 + z)*16 + c  ==  x<<18 | y<<11 | z<<4 | c
    const int xo0 = x0 << 18, xo1 = x1 << 18;
    const int yo0 = y0 << 11, yo1 = y1 << 11;
    const int zo0 = (z0 << 4) + c, zo1 = (z1 << 4) + c;

    // 8 corner gathers; each is a coalesced 64B request per half-wave,
    // L2-resident (128 MB grid < 192 MB L2).
    const float c000 = grid[xo0 + yo0 + zo0];
    const float c001 = grid[xo0 + yo0 + zo1];
    const float c010 = grid[xo0 + yo1 + zo0];
    const float c011 = grid[xo0 + yo1 + zo1];
    const float c100 = grid[xo1 + yo0 + zo0];
    const float c101 = grid[xo1 + yo0 + zo1];
    const float c110 = grid[xo1 + yo1 + zo0];
    const float c111 = grid[xo1 + yo1 + zo1];

    // 7-FMA lerp tree: a + t*(b - a).
    const float c00 = fmaf(xd, c100 - c000, c000);
    const float c01 = fmaf(xd, c101 - c001, c001);
    const float c10 = fmaf(xd, c110 - c010, c010);
    const float c11 = fmaf(xd, c111 - c011, c011);
    const float c0  = fmaf(yd, c10 - c00, c00);
    const float c1  = fmaf(yd, c11 - c01, c01);
    const float r   = fmaf(zd, c1 - c0, c0);

    // Streaming output: non-temporal so the 128 MB result doesn't evict the
    // grid from L2. Fully coalesced (lane stride 4B).
    __builtin_nontemporal_store(r, &out[(size_t)p * GC + c]);
}

extern "C" void kernel_launch(void* const* d_in, const int* in_sizes, int n_in,
                              void* d_out, int out_size, void* d_ws, size_t ws_size,
                              hipStream_t stream) {
    const float* pts  = (const float*)d_in[0];   // (N,3) f32
    const float* vals = (const float*)d_in[1];   // (128,128,128,16) f32
    float* out = (float*)d_out;                  // (N,16) f32

    const int npts    = in_sizes[0] / 3;
    const int nblocks = (npts + 15) / 16;        // 16 points per 256-thread block

    trilerp16_kernel<<<nblocks, 256, 0, stream>>>(pts, vals, out, npts);
}